// downprompt_61108794687801
// MI455X (gfx1250) — compile-verified
//
#include <hip/hip_runtime.h>
#include <cmath>

#define D512 512
#define NCLS 3

typedef __attribute__((ext_vector_type(2))) float v2f;
typedef __attribute__((ext_vector_type(8))) float v8f;

// Branchless ELU: hardware v_exp_f32 (TRANS pipe, co-executes with WMMA),
// no EXEC-mask branches. exp path only selected for x <= 0, and the fmin
// clamp keeps the exp argument <= 0 so there is no overflow concern.
__device__ __forceinline__ float eluf(float x) {
    float e = __expf(fminf(x, 0.0f)) - 1.0f;
    return x > 0.0f ? x : e;
}

// ---------------- kernel 0: zero the class-sum accumulators ----------------
__global__ void k_zero(float* __restrict__ sums) {
    int i = blockIdx.x * blockDim.x + threadIdx.x;
    if (i < NCLS * D512) sums[i] = 0.0f;
}

// ---------------- kernel 1: segment sum (labels -> [3,512]) ----------------
// blockDim = 256. Each thread owns columns t and t+256 of every class row in
// LDS, so no syncs / no conflicts; flush with global float atomics.
__global__ void k_segsum(const float* __restrict__ feature,
                         const float* __restrict__ seq,
                         const float* __restrict__ weight,
                         const int*   __restrict__ labels,
                         const int*   __restrict__ train,
                         float* __restrict__ sums,
                         int N, int rowsPerBlock) {
    __shared__ float acc[NCLS * D512];
    __shared__ float sw[D512];
    const int t = threadIdx.x;
    sw[t]       = weight[t];
    sw[t + 256] = weight[t + 256];
#pragma unroll
    for (int c = 0; c < NCLS; ++c) {
        acc[c * D512 + t]       = 0.0f;
        acc[c * D512 + t + 256] = 0.0f;
    }
    const int tr = train[0];
    int r0 = blockIdx.x * rowsPerBlock;
    int r1 = r0 + rowsPerBlock; if (r1 > N) r1 = N;
    for (int r = r0; r < r1; ++r) {
        const int lbl = labels[r];
        float v0, v1;
        if (tr == 1) {
            const float* s = seq + (size_t)r * D512;
            v0 = eluf(sw[t] * s[t]);
            v1 = eluf(sw[t + 256] * s[t + 256]);
        } else {
            const float* f = feature + (size_t)r * D512;
            v0 = f[t];
            v1 = f[t + 256];
        }
        acc[lbl * D512 + t]       += v0;
        acc[lbl * D512 + t + 256] += v1;
    }
#pragma unroll
    for (int c = 0; c < NCLS; ++c) {
        atomicAdd(&sums[c * D512 + t],       acc[c * D512 + t]);
        atomicAdd(&sums[c * D512 + t + 256], acc[c * D512 + t + 256]);
    }
}

// -------- kernel 2: ave = sums/(N//2), an[c] = ||ave[c]||  --------
// ws layout (floats): [0,1536) sums | [1536,3072) ave | [3072,3075) an
__global__ void k_finalize(float* __restrict__ ws, int N) {
    __shared__ float red[NCLS][256];
    const int t = threadIdx.x;
    const float inv = 1.0f / (float)(N / 2);
#pragma unroll
    for (int c = 0; c < NCLS; ++c) {
        float s0 = ws[c * D512 + t] * inv;
        float s1 = ws[c * D512 + t + 256] * inv;
        ws[1536 + c * D512 + t]       = s0;
        ws[1536 + c * D512 + t + 256] = s1;
        red[c][t] = s0 * s0 + s1 * s1;
    }
    __syncthreads();
    for (int s = 128; s > 0; s >>= 1) {
        if (t < s) {
#pragma unroll
            for (int c = 0; c < NCLS; ++c) red[c][t] += red[c][t + s];
        }
        __syncthreads();
    }
    if (t < NCLS) ws[3072 + t] = sqrtf(red[t][0]);
}

// -------- kernel 3: fused elu + row-norm + [16,512]x[512,16] WMMA + softmax -
// blockDim = 256 (8 waves, wave32). One wave owns a 16-row tile.
__global__ void k_main(const float* __restrict__ seq,
                       const float* __restrict__ weight,
                       const float* __restrict__ g_ave,
                       const float* __restrict__ g_an,
                       float* __restrict__ out,
                       int N) {
    __shared__ float s_ave[16 * D512];   // ave padded to 16 "classes" (32 KB)
    __shared__ float s_w[D512];          // 2 KB
    __shared__ float s_c[8][16 * 16];    // per-wave C spill (8 KB)

    const int t = threadIdx.x;
    for (int i = t; i < 16 * D512; i += 256)
        s_ave[i] = (i < NCLS * D512) ? g_ave[i] : 0.0f;
    s_w[t]       = weight[t];
    s_w[t + 256] = weight[t + 256];
    __syncthreads();

    const int lane = t & 31;
    const int wid  = t >> 5;
    const int ntiles = (N + 15) >> 4;
    const int tile = blockIdx.x * 8 + wid;
    if (tile >= ntiles) return;          // wave-uniform: EXEC stays all-ones

    const int m16 = lane & 15;           // A row index == B column index
    const int hk  = lane >> 4;           // which K-pair of the K=4 step
    int row = (tile << 4) + m16;
    const float rmask = (row < N) ? 1.0f : 0.0f;
    if (row >= N) row = N - 1;
    const float* srow = seq + (size_t)row * D512;
    const float* brow = s_ave + m16 * D512;

    v8f c = {};
    float ssq = 0.0f;
#pragma unroll 8
    for (int kb = 0; kb < D512; kb += 4) {
        const int k0 = kb + hk * 2;
        float2 sp = *(const float2*)(srow + k0);        // global_load_b64
        float2 wp = *(const float2*)(s_w + k0);         // ds_load_b64
        float2 bp = *(const float2*)(brow + k0);        // ds_load_b64
        float ax = eluf(sp.x * wp.x) * rmask;
        float ay = eluf(sp.y * wp.y) * rmask;
        ssq += ax * ax + ay * ay;
        v2f a; a.x = ax;   a.y = ay;
        v2f b; b.x = bp.x; b.y = bp.y;
        // D(16x16,f32) += A(16x4,f32) x B(4x16,f32)
        c = __builtin_amdgcn_wmma_f32_16x16x4_f32(
                false, a, false, b, (short)0, c, false, false);
    }

    // row sum-of-squares: combine the two K-half lanes (wave32 xor-16)
    const float ssqT = ssq + __shfl_xor(ssq, 16, 32);

    // spill C tile to LDS: VGPR v / lane L holds (M = v + 8*(L>=16), N = L&15)
    float* ct = &s_c[wid][0];
#pragma unroll
    for (int v = 0; v < 8; ++v)
        ct[(v + hk * 8) * 16 + m16] = c[v];
    asm volatile("s_wait_dscnt 0x0" ::: "memory");

    if (lane < 16) {
        const int orow = (tile << 4) + lane;
        if (orow < N) {
            const float rn = sqrtf(ssqT);
            const float d0 = ct[lane * 16 + 0];
            const float d1 = ct[lane * 16 + 1];
            const float d2 = ct[lane * 16 + 2];
            const float a0 = g_an[0], a1 = g_an[1], a2 = g_an[2];
            const float eps = 1e-8f;
            float s0 = d0 / fmaxf(rn * a0, eps);
            float s1 = d1 / fmaxf(rn * a1, eps);
            float s2 = d2 / fmaxf(rn * a2, eps);
            float mx = fmaxf(s0, fmaxf(s1, s2));
            float e0 = expf(s0 - mx), e1 = expf(s1 - mx), e2 = expf(s2 - mx);
            float is = 1.0f / (e0 + e1 + e2);
            out[(size_t)orow * 3 + 0] = e0 * is;
            out[(size_t)orow * 3 + 1] = e1 * is;
            out[(size_t)orow * 3 + 2] = e2 * is;
        }
    }
}

extern "C" void kernel_launch(void* const* d_in, const int* in_sizes, int n_in,
                              void* d_out, int out_size, void* d_ws, size_t ws_size,
                              hipStream_t stream) {
    (void)n_in; (void)out_size; (void)ws_size;
    const float* seq     = (const float*)d_in[0];
    const float* feature = (const float*)d_in[1];
    const int*   labels  = (const int*)d_in[2];
    const float* weight  = (const float*)d_in[3];
    const int*   train   = (const int*)d_in[4];
    float* out = (float*)d_out;
    float* ws  = (float*)d_ws;

    const int D = in_sizes[3];          // 512
    const int N = in_sizes[0] / D;      // 200000

    k_zero<<<(NCLS * D512 + 255) / 256, 256, 0, stream>>>(ws);

    const int B1 = 256;
    const int rowsPerBlock = (N + B1 - 1) / B1;
    k_segsum<<<B1, 256, 0, stream>>>(feature, seq, weight, labels, train,
                                     ws, N, rowsPerBlock);

    k_finalize<<<1, 256, 0, stream>>>(ws, N);

    const int ntiles = (N + 15) / 16;
    const int B2 = (ntiles + 7) / 8;
    k_main<<<B2, 256, 0, stream>>>(seq, weight, ws + 1536, ws + 3072, out, N);
}